// SampleLayer_84507776516247
// MI455X (gfx1250) — compile-verified
//
#include <hip/hip_runtime.h>
#include <hip/hip_bf16.h>
#include <stdint.h>

// ---------------- problem constants (from setup_inputs) ----------------
#define NBATCH 8
#define NPTS   4096
#define SSEL   1024                    // num_to_sample
#define EEXC   102                     // int(0.1 * SSEL)

// ---------------- FPS kernel config ----------------
#define TA           512               // threads per block = 16 wave32
#define WAVES_A      (TA / 32)         // 16
#define PTS_PER_WAVE (NPTS / WAVES_A)  // 256
#define TILES        (PTS_PER_WAVE / 16) // 16 tiles of 16 points per wave

typedef float        v2f   __attribute__((ext_vector_type(2)));
typedef float        v8f   __attribute__((ext_vector_type(8)));
typedef unsigned int u32x4 __attribute__((ext_vector_type(4)));
typedef int          i32x4 __attribute__((ext_vector_type(4)));
typedef int          i32x8 __attribute__((ext_vector_type(8)));

// =====================================================================
// Kernel A: farthest-point-sampling order, one workgroup per batch.
// Points live in LDS (48KB), staged by the Tensor Data Mover.
// Distance update per step runs on the matrix pipe:
//   D = A x B + C with A rows = (-2lx,-2ly,-2lz,1), B cols = (x,y,z,|p|^2),
//   C = |lp|^2  ==>  D[i][j] = d^2(p_j, last)  exactly, fp32.
// =====================================================================
__global__ void __launch_bounds__(TA)
fps_kernel(const float* __restrict__ x, int* __restrict__ order) {
  __shared__ float pts[NPTS * 3];     // AoS xyz for this batch (48KB)
  __shared__ float redV[WAVES_A];
  __shared__ int   redI[WAVES_A];
  __shared__ int   bidxS;

  const int b    = blockIdx.x;
  const int tid  = threadIdx.x;
  const int lane = tid & 31;
  const int wave = tid >> 5;
  const float* xb = x + (size_t)b * NPTS * 3;

  // ---- stage the 48KB point tile into LDS ----
#if __has_builtin(__builtin_amdgcn_tensor_load_to_lds) && __has_builtin(__builtin_amdgcn_s_wait_tensorcnt)
  if (wave == 0) {  // one TDM issue per workgroup (EXEC ignored by TDM)
    const uint32_t ldsAddr = (uint32_t)(uintptr_t)&pts[0]; // low 32b of generic ptr = LDS offset
    const uint64_t ga      = (uint64_t)(uintptr_t)xb;
    const uint32_t elems   = NPTS * 3;                     // 12288 x f32, 1-D copy
    u32x4 g0; i32x8 g1; i32x4 g2; i32x4 g3;
    // D# group0: count=1, lds_addr, global_addr[56:0], type=2 ("image")
    g0[0] = 1u;
    g0[1] = ldsAddr;
    g0[2] = (uint32_t)ga;
    g0[3] = (uint32_t)((ga >> 32) & 0x01FFFFFFull) | (2u << 30);
    // D# group1: wg_mask=0, data_size=2 (4B); tensor_dim0=tile_dim0=12288,
    // tensor_dim1=tile_dim1=1, dim0_stride=12288
    g1[0] = (int)(2u << 16);                               // data_size=4B
    g1[1] = (int)((elems & 0xFFFFu) << 16);                // tensor_dim0[15:0]
    g1[2] = (int)(((elems >> 16) & 0xFFFFu) | (1u << 16)); // dim0 hi | tensor_dim1=1
    g1[3] = (int)((elems & 0xFFFFu) << 16);                // tile_dim0
    g1[4] = 1;                                             // tile_dim1=1, tile_dim2=0
    g1[5] = (int)elems;                                    // tensor_dim0_stride[31:0]
    g1[6] = 0; g1[7] = 0;
    g2[0] = g2[1] = g2[2] = g2[3] = 0;
    g3[0] = g3[1] = g3[2] = g3[3] = 0;
#if __clang_major__ >= 23
    i32x8 g4; g4[0]=g4[1]=g4[2]=g4[3]=g4[4]=g4[5]=g4[6]=g4[7]=0;
    __builtin_amdgcn_tensor_load_to_lds(g0, g1, g2, g3, g4, 0);
#else
    __builtin_amdgcn_tensor_load_to_lds(g0, g1, g2, g3, 0);
#endif
    __builtin_amdgcn_s_wait_tensorcnt(0);
  }
#else
  for (int i = tid; i < NPTS * 3; i += TA) pts[i] = xb[i];
#endif
  if (tid == 0) { bidxS = 0; order[(size_t)b * NPTS] = 0; }
  __syncthreads();

#if __has_builtin(__builtin_amdgcn_wmma_f32_16x16x4_f32)
  // Persistent B tiles: 4x16 f32 B layout -> 2 VGPRs/tile:
  //   VGPR0: lanes 0-15 = row K0 (x), lanes 16-31 = row K1 (y)
  //   VGPR1: lanes 0-15 = row K2 (z), lanes 16-31 = row K3 (|p|^2)
  const bool hi = (lane >= 16);
  const int  pl = lane & 15;
  float bm0[TILES], bm1[TILES], mind[TILES];
#pragma unroll
  for (int t = 0; t < TILES; ++t) {
    const int p  = wave * PTS_PER_WAVE + t * 16 + pl;
    const float px = pts[3 * p + 0];
    const float py = pts[3 * p + 1];
    const float pz = pts[3 * p + 2];
    const float nn = px * px + py * py + pz * pz;
    bm0[t] = hi ? py : px;
    bm1[t] = hi ? nn : pz;
    mind[t] = 1e30f;
  }
#else
  constexpr int PPT = NPTS / TA;      // scalar-VALU fallback: 8 points/thread
  float mind[PPT];
#pragma unroll
  for (int q = 0; q < PPT; ++q) mind[q] = 1e30f;
#endif

  for (int it = 1; it < NPTS; ++it) {
    const int   lastp = bidxS;        // uniform LDS broadcast
    const float lx = pts[3 * lastp + 0];
    const float ly = pts[3 * lastp + 1];
    const float lz = pts[3 * lastp + 2];

    float bestv = -1.0f;
    int   besti = 0;

#if __has_builtin(__builtin_amdgcn_wmma_f32_16x16x4_f32)
    const float ln = lx * lx + ly * ly + lz * lz;
    // A: 16x4, all rows = (-2lx,-2ly,-2lz,1). A layout: lanes0-15 K0/K1, lanes16-31 K2/K3
    v2f amat;
    amat.x = hi ? (-2.0f * lz) : (-2.0f * lx);
    amat.y = hi ? 1.0f         : (-2.0f * ly);
    v8f cmat;
#pragma unroll
    for (int e = 0; e < 8; ++e) cmat[e] = ln;   // C = |lp|^2 everywhere
#pragma unroll
    for (int t = 0; t < TILES; ++t) {
      v2f bmat; bmat.x = bm0[t]; bmat.y = bm1[t];
      v8f d = __builtin_amdgcn_wmma_f32_16x16x4_f32(
          false, amat, false, bmat, (short)0, cmat, false, false);
      // every row of D holds d^2 of column j; lane l owns column l&15
      const float mm = fminf(mind[t], d[0]);
      mind[t] = mm;
      if (mm > bestv) {               // t ascending => first-max (lowest idx) kept
        bestv = mm;
        besti = wave * PTS_PER_WAVE + t * 16 + pl;
      }
    }
#else
#pragma unroll
    for (int q = 0; q < PPT; ++q) {
      const int p  = tid * PPT + q;
      const float dx = pts[3 * p + 0] - lx;
      const float dy = pts[3 * p + 1] - ly;
      const float dz = pts[3 * p + 2] - lz;
      const float mm = fminf(mind[q], dx * dx + dy * dy + dz * dz);
      mind[q] = mm;
      if (mm > bestv) { bestv = mm; besti = p; }
    }
#endif

    // wave32 argmax reduction (first-max tie-break: lower index wins)
#pragma unroll
    for (int off = 16; off > 0; off >>= 1) {
      const float ov = __shfl_xor(bestv, off, 32);
      const int   oi = __shfl_xor(besti, off, 32);
      if (ov > bestv || (ov == bestv && oi < besti)) { bestv = ov; besti = oi; }
    }
    if (lane == 0) { redV[wave] = bestv; redI[wave] = besti; }
    __syncthreads();
    if (tid == 0) {
      float bv = redV[0]; int bi = redI[0];
      for (int w = 1; w < WAVES_A; ++w) {
        if (redV[w] > bv || (redV[w] == bv && redI[w] < bi)) { bv = redV[w]; bi = redI[w]; }
      }
      bidxS = bi;
      order[(size_t)b * NPTS + it] = bi;
    }
    __syncthreads();
  }
}

// =====================================================================
// Kernel B: curvature normalization, sc, two bitonic sorts, merge, gather
// =====================================================================
__device__ inline void bitonic_sort(float* key, int* idx, int n, bool asc, int tid) {
  for (int k = 2; k <= n; k <<= 1) {
    for (int j = k >> 1; j > 0; j >>= 1) {
      for (int i = tid; i < n; i += 1024) {
        const int ixj = i ^ j;
        if (ixj > i) {
          const float a  = key[i], c = key[ixj];
          const int   ia = idx[i], ic = idx[ixj];
          // total order: (key asc|desc), ties broken by ascending index
          const bool ji_less = asc ? (c < a || (c == a && ic < ia))
                                   : (c > a || (c == a && ic < ia));
          const bool ij_less = asc ? (a < c || (a == c && ia < ic))
                                   : (a > c || (a == c && ia < ic));
          const bool sw = ((i & k) == 0) ? ji_less : ij_less;
          if (sw) { key[i] = c; key[ixj] = a; idx[i] = ic; idx[ixj] = ia; }
        }
      }
      __syncthreads();
    }
  }
}

__global__ void __launch_bounds__(1024)
select_kernel(const float* __restrict__ x, const float* __restrict__ curv,
              const int* __restrict__ order, float* __restrict__ out) {
  __shared__ float skey[NPTS];
  __shared__ int   sidx[NPTS];
  __shared__ float sc[NPTS];
  __shared__ float baK[SSEL];
  __shared__ int   baI[SSEL];
  __shared__ float redM[32], redX[32];

  const int b = blockIdx.x, tid = threadIdx.x;
  const int lane = tid & 31, wave = tid >> 5;
  const float* cb = curv + (size_t)b * NPTS;
  const int*   ob = order + (size_t)b * NPTS;

  // per-batch curv min / max
  float mn = 1e30f, mx = -1e30f;
  for (int i = tid; i < NPTS; i += 1024) {
    const float v = cb[i];
    mn = fminf(mn, v); mx = fmaxf(mx, v);
  }
#pragma unroll
  for (int off = 16; off > 0; off >>= 1) {
    mn = fminf(mn, __shfl_xor(mn, off, 32));
    mx = fmaxf(mx, __shfl_xor(mx, off, 32));
  }
  if (lane == 0) { redM[wave] = mn; redX[wave] = mx; }
  __syncthreads();
  if (tid == 0) {
    float a = redM[0], c = redX[0];
    for (int w = 1; w < 32; ++w) { a = fminf(a, redM[w]); c = fmaxf(c, redX[w]); }
    redM[0] = a; redX[0] = c;
  }
  __syncthreads();
  mn = redM[0]; mx = redX[0];
  const float inv = 1.0f / (mx - mn);

  // sc[i] = linspace(1,0,N)[idxs[i]] * curvn[i]
  for (int i = tid; i < NPTS; i += 1024) {
    const int   j   = ob[i];
    const float lin = 1.0f - (float)j * (1.0f / (float)(NPTS - 1));
    sc[i] = lin * (cb[i] - mn) * inv;
  }
  __syncthreads();

  // sort A: sc[0:S] descending (== top_k(sca, S))
  skey[tid] = sc[tid]; sidx[tid] = tid;
  __syncthreads();
  bitonic_sort(skey, sidx, SSEL, false, tid);
  baK[tid] = skey[tid]; baI[tid] = sidx[tid];
  __syncthreads();

  // sort B: sc[S:N] (3072) padded to 4096 with +inf, ascending;
  // top-m largest in ascending order sit at positions [2048, 3072)
  for (int i = tid; i < NPTS; i += 1024) {
    if (i < NPTS - SSEL) { skey[i] = sc[SSEL + i]; sidx[i] = SSEL + i; }
    else                 { skey[i] = 3.0e38f;      sidx[i] = 0x7FFFFFFF; }
  }
  __syncthreads();
  bitonic_sort(skey, sidx, NPTS, true, tid);

  // merge rule (algebraically simplified reference):
  //   j <  S-E : final = ba_idx[j]
  //   j >= S-E : final = tb_idx[j] if tb_vals[j] > ba_vals[j] else ba_idx[j]
  const float bav = baK[tid]; const int bai = baI[tid];
  const float tbv = skey[2048 + tid]; const int tbi = sidx[2048 + tid];
  int fin = bai;
  if (tid >= (SSEL - EEXC) && tbv > bav) fin = tbi;

  out[(size_t)b * SSEL + tid] = (float)fin;          // output 0: final (as float)
  const int pt = ob[fin];                            // pt_idx = idxs[final]
  const float* xp = x + ((size_t)b * NPTS + pt) * 3;
  float* op = out + (size_t)NBATCH * SSEL + ((size_t)b * SSEL + tid) * 3;
  op[0] = xp[0]; op[1] = xp[1]; op[2] = xp[2];       // output 1: sampled points
}

// =====================================================================
extern "C" void kernel_launch(void* const* d_in, const int* in_sizes, int n_in,
                              void* d_out, int out_size, void* d_ws, size_t ws_size,
                              hipStream_t stream) {
  const float* x    = (const float*)d_in[0];   // (8,4096,3) f32
  const float* curv = (const float*)d_in[1];   // (8,4096)   f32
  int*   order = (int*)d_ws;                   // 8*4096 ints = 128KB scratch
  float* out   = (float*)d_out;                // 8192 final + 24576 sampled

  fps_kernel<<<NBATCH, TA, 0, stream>>>(x, order);
  select_kernel<<<NBATCH, 1024, 0, stream>>>(x, curv, order, out);

  (void)in_sizes; (void)n_in; (void)out_size; (void)ws_size;
}